// MixtureExperts_64390149701863
// MI455X (gfx1250) — compile-verified
//
#include <hip/hip_runtime.h>

// MoE score-weighted prompt mixture as a dense GEMM on the f32 WMMA pipe:
//   out[2048 x 16384] = S[2048 x 64] * W[64 x 16384]
// where S is the scatter-add-densified routing matrix.
// Memory-bound: 128MB NT output stream (~5.5us @ 23.3TB/s); W (4MB) stays in L2.

typedef __attribute__((ext_vector_type(2))) float v2f;
typedef __attribute__((ext_vector_type(8))) float v8f;

#define MOE_BS        2048
#define MOE_EXPERTS   64
#define MOE_TOPK      8
#define MOE_COLS      16384   // PROMPT_LEN * DIM = 32 * 512
#define LDS_STRIDE    65      // 64 + 1 pad -> conflict-free A-fragment reads

__launch_bounds__(256, 4)
__global__ void moe_mix_wmma_f32(const float* __restrict__ score,
                                 const int*   __restrict__ idx,
                                 const float* __restrict__ W,
                                 float*       __restrict__ out) {
    __shared__ float sS[16 * LDS_STRIDE];   // 16 tokens x 64 experts (padded)

    const int tid = threadIdx.x;
    const int m0  = blockIdx.y << 4;        // first token of this M-tile

    // ---- Stage 1: densify routing scores into LDS (zero + scatter-add) ----
    for (int i = tid; i < 16 * LDS_STRIDE; i += 256) sS[i] = 0.0f;
    __syncthreads();
    if (tid < 16) {                         // one thread per token row: no atomics,
        const int b = m0 + tid;             // duplicate experts accumulate correctly
        float* row = &sS[tid * LDS_STRIDE];
        #pragma unroll
        for (int k = 0; k < MOE_TOPK; ++k) {
            const int e = idx[b * MOE_TOPK + k];
            row[e] += score[b * MOE_EXPERTS + e];
        }
    }
    __syncthreads();

    // ---- Stage 2: 8 waves x one 16-col N-tile, K=64 via 16 x wmma 16x16x4 ----
    const int wave = tid >> 5;
    const int lane = tid & 31;
    const int half = lane >> 4;             // 0: lanes 0-15, 1: lanes 16-31
    const int l15  = lane & 15;
    const int n    = (blockIdx.x << 7) + (wave << 4) + l15;  // output column

    const float* __restrict__ srow = &sS[l15 * LDS_STRIDE];  // A row for this lane
    v8f acc = {};

    #pragma unroll
    for (int k = 0; k < 16; ++k) {
        // element (vgpr v, half h) <-> K = 4k + 2h + v   (ISA 7.12.2, f32 16x4 A)
        const int kc = (k << 2) + (half << 1);
        v2f a, b;
        a.x = srow[kc];
        a.y = srow[kc + 1];
        b.x = W[(size_t)kc       * MOE_COLS + n];   // L2-resident weight rows
        b.y = W[(size_t)(kc + 1) * MOE_COLS + n];
        acc = __builtin_amdgcn_wmma_f32_16x16x4_f32(
            /*neg_a=*/false, a, /*neg_b=*/false, b,
            /*c_mod=*/(short)0, acc, /*reuse_a=*/false, /*reuse_b=*/false);
    }

    // ---- Stage 3: NT store (don't evict the L2-resident weight table) ----
    #pragma unroll
    for (int v = 0; v < 8; ++v) {
        const int m = m0 + v + (half << 3);          // C/D layout: vgpr r -> row r+8h
        __builtin_nontemporal_store(acc[v], &out[(size_t)m * MOE_COLS + n]);
    }
}

extern "C" void kernel_launch(void* const* d_in, const int* in_sizes, int n_in,
                              void* d_out, int out_size, void* d_ws, size_t ws_size,
                              hipStream_t stream) {
    const float* score = (const float*)d_in[0];   // [2048, 64] f32
    const int*   idx   = (const int*)  d_in[1];   // [2048, 8]  int
    const float* W     = (const float*)d_in[2];   // [64, 32, 512] f32 -> [64][16384]
    float*       out   = (float*)d_out;           // [2048, 32, 512] f32

    dim3 grid(MOE_COLS / 128, MOE_BS / 16);       // 128 N-blocks x 128 M-tiles
    dim3 block(256);                              // 8 waves (wave32)
    moe_mix_wmma_f32<<<grid, block, 0, stream>>>(score, idx, W, out);
}